// HyperUNet_87978110091287
// MI455X (gfx1250) — compile-verified
//
#include <hip/hip_runtime.h>
#include <math.h>

typedef __attribute__((ext_vector_type(2))) float v2f;
typedef __attribute__((ext_vector_type(8))) float v8f;

#define NFULL 8192
#define DFEAT 128

// ---------------------------------------------------------------------------
// CDNA5 async global->LDS copy (16 bytes per lane), tracked by ASYNCcnt.
// vdst = 32-bit LDS byte address, vaddr = 64-bit global address.
// ---------------------------------------------------------------------------
__device__ __forceinline__ void async_copy_b128(unsigned lds_off, const float* gaddr) {
  asm volatile("global_load_async_to_lds_b128 %0, %1, off"
               :: "v"(lds_off), "v"(gaddr)
               : "memory");
}

// ---------------------------------------------------------------------------
// score = tanh((x . w) / ||w||), one wave32 per row
// ---------------------------------------------------------------------------
__global__ void score_kernel(const float* __restrict__ x, const float* __restrict__ w,
                             float* __restrict__ score, int n) {
  int gwave = (int)((blockIdx.x * blockDim.x + threadIdx.x) >> 5);
  int lane = threadIdx.x & 31;
  if (gwave >= n) return;
  const float* xr = x + (size_t)gwave * DFEAT;
  float ws = 0.f, dot = 0.f;
  for (int c = lane; c < DFEAT; c += 32) {
    float wv = w[c];
    ws += wv * wv;
    dot += xr[c] * wv;
  }
  for (int off = 16; off > 0; off >>= 1) {
    ws  += __shfl_xor(ws, off, 32);
    dot += __shfl_xor(dot, off, 32);
  }
  if (lane == 0) score[gwave] = tanhf(dot / sqrtf(ws));
}

// ---------------------------------------------------------------------------
// Exact top-k via full bitonic sort in LDS (n power of two, n <= 8192).
// Descending by score, ascending index on ties (matches jax.lax.top_k).
// ---------------------------------------------------------------------------
__global__ __launch_bounds__(1024) void topk_kernel(const float* __restrict__ score,
                                                    int n, int k,
                                                    int* __restrict__ perm,
                                                    float* __restrict__ topsc) {
  __shared__ float key[8192];
  __shared__ unsigned short idx[8192];
  int tid = threadIdx.x;
  for (int i = tid; i < n; i += 1024) { key[i] = score[i]; idx[i] = (unsigned short)i; }
  __syncthreads();
  for (int size = 2; size <= n; size <<= 1) {
    for (int stride = size >> 1; stride > 0; stride >>= 1) {
      for (int t = tid; t < (n >> 1); t += 1024) {
        int i = ((t / stride) * (stride << 1)) + (t & (stride - 1));
        int j = i + stride;
        bool desc = ((i & size) == 0);
        float ki = key[i], kj = key[j];
        unsigned short ii = idx[i], ij = idx[j];
        bool bigger_j = (kj > ki) || (kj == ki && ij < ii);
        if (bigger_j == desc) {
          key[i] = kj; key[j] = ki; idx[i] = ij; idx[j] = ii;
        }
      }
      __syncthreads();
    }
  }
  for (int i = tid; i < k; i += 1024) { perm[i] = (int)idx[i]; topsc[i] = key[i]; }
}

// ---------------------------------------------------------------------------
// out[b,:] = x[perm[b],:] * sc[b]
// ---------------------------------------------------------------------------
__global__ void gather_gate_kernel(const float* __restrict__ x, const int* __restrict__ perm,
                                   const float* __restrict__ sc, float* __restrict__ out) {
  int b = blockIdx.x, t = threadIdx.x;
  out[(size_t)b * DFEAT + t] = x[(size_t)perm[b] * DFEAT + t] * sc[b];
}

// ---------------------------------------------------------------------------
// Hg[i,j] = H[perm[i], perm[j]]   (full H is NFULL wide)
// ---------------------------------------------------------------------------
__global__ void gather_h_kernel(const float* __restrict__ H, const int* __restrict__ perm,
                                float* __restrict__ Hg, int k) {
  int row = blockIdx.y;
  int j = blockIdx.x * blockDim.x + threadIdx.x;
  if (j >= k) return;
  Hg[(size_t)row * k + j] = H[(size_t)perm[row] * NFULL + perm[j]];
}

__global__ void copy_kernel(const float* __restrict__ src, float* __restrict__ dst, size_t n) {
  size_t i = blockIdx.x * (size_t)blockDim.x + threadIdx.x;
  if (i < n) dst[i] = src[i];
}

// out[perm[b],:] += x[b,:]  (perm entries unique)
__global__ void scatter_add_kernel(float* __restrict__ out, const int* __restrict__ perm,
                                   const float* __restrict__ x) {
  int b = blockIdx.x, t = threadIdx.x;
  out[(size_t)perm[b] * DFEAT + t] += x[(size_t)b * DFEAT + t];
}

// ---------------------------------------------------------------------------
// C[M x 128] = (relu?)( A[M x Kdim] @ B[Kdim x 128] + bias )
// f32 WMMA 16x16x4.  Block = 8 waves; block owns 16 rows x 128 cols so A
// streams from HBM exactly once.  A chunk (16 x 64) is DMA'd to LDS with
// global_load_async_to_lds_b128 (ASYNCcnt), double-buffered so the next
// chunk's DMA overlaps the current chunk's WMMAs.
// LDS row stride 68 floats: 16B-aligned B128 destinations AND conflict-free
// column reads (68 mod 64 = 4 -> 16 distinct banks per half-wave).
// Fragment layouts per ISA 7.12.2:
//   A 16x4 : lane<16 -> row=lane, K = {0,1}; lane>=16 -> row=lane-16, K={2,3}
//   B 4x16 : lane<16 -> col=lane, K = {0,1}; lane>=16 -> col=lane-16, K={2,3}
//   C/D    : VGPR v  : lane<16 -> row v, col=lane; lane>=16 -> row v+8
// ---------------------------------------------------------------------------
#define ASTRIDE 68
__global__ __launch_bounds__(256) void gemm_wmma_kernel(
    const float* __restrict__ A, int lda,
    const float* __restrict__ B,
    const float* __restrict__ bias,   // may be null
    float* __restrict__ C,
    int Kdim, int use_relu) {
  __shared__ float As[2 * 16 * ASTRIDE];
  int tid  = threadIdx.x;
  int wave = tid >> 5;
  int lane = tid & 31;
  int r0 = blockIdx.x * 16;
  int c0 = wave * 16;
  int lcol = lane & 15;
  int col = c0 + lcol;
  int khalf = (lane < 16) ? 0 : 2;

  // per-thread staging coordinates (16 rows x 64 floats per chunk, b128 each)
  int sr  = tid >> 4;           // 0..15
  int sc4 = (tid & 15) << 2;    // 0..60
  const float* gsrc = A + (size_t)(r0 + sr) * lda + sc4;
  unsigned lds_dst0 = (unsigned)(size_t)(&As[sr * ASTRIDE + sc4]);
  const unsigned bufbytes = 16 * ASTRIDE * 4;

  v8f acc;
  float binit = bias ? bias[col] : 0.0f;
  for (int v = 0; v < 8; ++v) acc[v] = binit;

  // prologue: DMA chunk 0 into buffer 0
  async_copy_b128(lds_dst0, gsrc);

  int nchunks = Kdim >> 6;
  for (int c = 0; c < nchunks; ++c) {
    int cur = c & 1;
    int kk0 = c << 6;
    if (c + 1 < nchunks) {
      // launch DMA of next chunk into the other buffer, then wait for current
      async_copy_b128(lds_dst0 + (cur ^ 1) * bufbytes, gsrc + (size_t)(c + 1) * 64);
      __builtin_prefetch(B + (size_t)(kk0 + 64) * DFEAT + col, 0, 3);
      asm volatile("s_wait_asynccnt 0x1" ::: "memory");
    } else {
      asm volatile("s_wait_asynccnt 0x0" ::: "memory");
    }
    __syncthreads();  // all waves' DMA for chunk c complete -> LDS valid

    const float* abase = &As[cur * 16 * ASTRIDE];
    for (int kk = 0; kk < 64; kk += 4) {
      v2f a, b;
      a.x = abase[lcol * ASTRIDE + kk + khalf + 0];
      a.y = abase[lcol * ASTRIDE + kk + khalf + 1];
      const float* bp = B + (size_t)(kk0 + kk + khalf) * DFEAT + col;
      b.x = bp[0];
      b.y = bp[DFEAT];
      acc = __builtin_amdgcn_wmma_f32_16x16x4_f32(false, a, false, b,
                                                  (short)0, acc, false, false);
    }
    __syncthreads();  // done reading this buffer before it is overwritten
  }

  int mbase = (lane < 16) ? 0 : 8;
  for (int v = 0; v < 8; ++v) {
    float val = acc[v];
    if (use_relu) val = fmaxf(val, 0.0f);
    C[(size_t)(r0 + mbase + v) * DFEAT + col] = val;
  }
}

// ---------------------------------------------------------------------------
extern "C" void kernel_launch(void* const* d_in, const int* in_sizes, int n_in,
                              void* d_out, int out_size, void* d_ws, size_t ws_size,
                              hipStream_t stream) {
  (void)in_sizes; (void)n_in; (void)out_size; (void)ws_size;
  const float* feat = (const float*)d_in[0];
  const float* H    = (const float*)d_in[1];
  const float* pw0  = (const float*)d_in[2];
  const float* pw1  = (const float*)d_in[3];
  const float* pw2  = (const float*)d_in[4];
  const float* Wd0  = (const float*)d_in[5];
  const float* bd0  = (const float*)d_in[6];
  const float* Wd1  = (const float*)d_in[7];
  const float* bd1  = (const float*)d_in[8];
  const float* Wd2  = (const float*)d_in[9];
  const float* bd2  = (const float*)d_in[10];
  const float* Wu0  = (const float*)d_in[11];
  const float* bu0  = (const float*)d_in[12];
  const float* Wu1  = (const float*)d_in[13];
  const float* bu1  = (const float*)d_in[14];
  const float* Wu2  = (const float*)d_in[15];
  const float* bu2  = (const float*)d_in[16];
  float* out = (float*)d_out;

  const int n0 = 8192, k0 = 4096;
  const int n1 = 4096, k1 = 2048;
  const int n2 = 2048, k2 = 1024;

  // carve workspace
  char* wsp = (char*)d_ws;
  size_t off = 0;
  auto carve = [&](size_t bytes) {
    char* p = wsp + off;
    off = (off + bytes + 255) & ~(size_t)255;
    return p;
  };
  float* scores = (float*)carve(8192 * 4);
  float* sc0    = (float*)carve((size_t)k0 * 4);
  float* sc1    = (float*)carve((size_t)k1 * 4);
  float* sc2    = (float*)carve((size_t)k2 * 4);
  int*   perm0  = (int*)carve((size_t)k0 * 4);
  int*   perm1  = (int*)carve((size_t)k1 * 4);
  int*   perm2  = (int*)carve((size_t)k2 * 4);
  float* xa  = (float*)carve((size_t)n0 * DFEAT * 4);
  float* xb  = (float*)carve((size_t)n0 * DFEAT * 4);
  float* xs1 = (float*)carve((size_t)k0 * DFEAT * 4);   // x after down conv 0
  float* xs2 = (float*)carve((size_t)k1 * DFEAT * 4);   // x after down conv 1
  float* H0g = (float*)carve((size_t)k0 * k0 * 4);
  float* H1g = (float*)carve((size_t)k1 * k1 * 4);
  float* H2g = (float*)carve((size_t)k2 * k2 * 4);

  // ---------------- down level 0 ----------------
  score_kernel<<<n0 * 32 / 256, 256, 0, stream>>>(feat, pw0, scores, n0);
  topk_kernel<<<1, 1024, 0, stream>>>(scores, n0, k0, perm0, sc0);
  gather_gate_kernel<<<k0, 128, 0, stream>>>(feat, perm0, sc0, xa);
  gather_h_kernel<<<dim3(k0 / 256, k0), 256, 0, stream>>>(H, perm0, H0g, k0);
  gemm_wmma_kernel<<<k0 / 16, 256, 0, stream>>>(xa, DFEAT, Wd0, bd0, xb, DFEAT, 0);
  gemm_wmma_kernel<<<k0 / 16, 256, 0, stream>>>(H0g, k0, xb, nullptr, xs1, k0, 1);

  // ---------------- down level 1 ----------------
  score_kernel<<<n1 * 32 / 256, 256, 0, stream>>>(xs1, pw1, scores, n1);
  topk_kernel<<<1, 1024, 0, stream>>>(scores, n1, k1, perm1, sc1);
  gather_gate_kernel<<<k1, 128, 0, stream>>>(xs1, perm1, sc1, xa);
  gather_h_kernel<<<dim3(k1 / 256, k1), 256, 0, stream>>>(H, perm1, H1g, k1);
  gemm_wmma_kernel<<<k1 / 16, 256, 0, stream>>>(xa, DFEAT, Wd1, bd1, xb, DFEAT, 0);
  gemm_wmma_kernel<<<k1 / 16, 256, 0, stream>>>(H1g, k1, xb, nullptr, xs2, k1, 1);

  // ---------------- down level 2 ----------------
  score_kernel<<<n2 * 32 / 256, 256, 0, stream>>>(xs2, pw2, scores, n2);
  topk_kernel<<<1, 1024, 0, stream>>>(scores, n2, k2, perm2, sc2);
  gather_gate_kernel<<<k2, 128, 0, stream>>>(xs2, perm2, sc2, xa);
  gather_h_kernel<<<dim3(k2 / 256, k2), 256, 0, stream>>>(H, perm2, H2g, k2);
  gemm_wmma_kernel<<<k2 / 16, 256, 0, stream>>>(xa, DFEAT, Wd2, bd2, xb, DFEAT, 0);
  gemm_wmma_kernel<<<k2 / 16, 256, 0, stream>>>(H2g, k2, xb, nullptr, xa, k2, 1);
  // xa = bottom x (k2 x 128)

  // ---------------- up level 0 (j=2) ----------------
  copy_kernel<<<(int)(((size_t)n2 * DFEAT + 255) / 256), 256, 0, stream>>>(xs2, xb, (size_t)n2 * DFEAT);
  scatter_add_kernel<<<k2, 128, 0, stream>>>(xb, perm2, xa);
  gemm_wmma_kernel<<<n2 / 16, 256, 0, stream>>>(xb, DFEAT, Wu0, bu0, xa, DFEAT, 0);
  gemm_wmma_kernel<<<n2 / 16, 256, 0, stream>>>(H1g, n2, xa, nullptr, xb, n2, 1);
  // xb = x (n2 x 128)

  // ---------------- up level 1 (j=1) ----------------
  copy_kernel<<<(int)(((size_t)n1 * DFEAT + 255) / 256), 256, 0, stream>>>(xs1, xa, (size_t)n1 * DFEAT);
  scatter_add_kernel<<<k1, 128, 0, stream>>>(xa, perm1, xb);
  gemm_wmma_kernel<<<n1 / 16, 256, 0, stream>>>(xa, DFEAT, Wu1, bu1, xb, DFEAT, 0);
  gemm_wmma_kernel<<<n1 / 16, 256, 0, stream>>>(H0g, n1, xb, nullptr, xa, n1, 1);
  // xa = x (n1 x 128)

  // ---------------- up level 2 (j=0) ----------------
  copy_kernel<<<(int)(((size_t)n0 * DFEAT + 255) / 256), 256, 0, stream>>>(feat, xb, (size_t)n0 * DFEAT);
  scatter_add_kernel<<<k0, 128, 0, stream>>>(xb, perm0, xa);
  gemm_wmma_kernel<<<n0 / 16, 256, 0, stream>>>(xb, DFEAT, Wu2, bu2, xa, DFEAT, 0);
  gemm_wmma_kernel<<<n0 / 16, 256, 0, stream>>>(H, NFULL, xa, nullptr, out, n0, 1);
}